// VGANet_53163105190000
// MI455X (gfx1250) — compile-verified
//
#include <hip/hip_runtime.h>
#include <hip/hip_bf16.h>
#include <stdint.h>

#define N_NODES 8192
#define F_IN    512
#define H_DIM   256
#define L_DIM   64
#define N_EDGES 262144

typedef float v2f __attribute__((ext_vector_type(2)));
typedef float v8f __attribute__((ext_vector_type(8)));

// ---------------------------------------------------------------------------
// Degree / normalization
// ---------------------------------------------------------------------------
__global__ void deg_init_kernel(float* __restrict__ deg) {
    int i = blockIdx.x * blockDim.x + threadIdx.x;
    deg[i] = 1.0f;                       // self-loop contribution of A + I
}

__global__ void deg_accum_kernel(const long long* __restrict__ ei,
                                 float* __restrict__ deg) {
    int e = blockIdx.x * blockDim.x + threadIdx.x;
    int d = (int)ei[N_EDGES + e];        // dst row of edge_index
    atomicAdd(&deg[d], 1.0f);
}

__global__ void dinv_kernel(const float* __restrict__ deg,
                            float* __restrict__ dinv) {
    int i = blockIdx.x * blockDim.x + threadIdx.x;
    dinv[i] = rsqrtf(deg[i]);
}

// ---------------------------------------------------------------------------
// Generic fp32 WMMA GEMM:  Craw = A[M,K] @ B[K,N]
//                          Cinit = Craw * dinv[row]^2   (self-loop init)
// Block: 256 threads = 8 waves. Block tile 128x64, wave w owns rows 16w..16w+15
// and all 64 cols via 4 accumulators (16x16 each).
// f32 WMMA fragment layout (ISA 7.12.2):
//   A 16x4 : lane r (0-15)/half h : a[0]=A[m0+r][k0+2h], a[1]=A[m0+r][k0+2h+1]
//   B 4x16 : b[0]=B[k0+2h][n0+r],  b[1]=B[k0+2h+1][n0+r]
//   C 16x16: c[v] = C[m0+v+8h][n0+r]
// B (the weight matrix, <=512KB) is L2/WGP$-resident; direct loads pipeline
// fine (verified in round-1 asm).
// ---------------------------------------------------------------------------
__global__ __launch_bounds__(256) void wmma_gemm_f32_kernel(
    const float* __restrict__ A, const float* __restrict__ B,
    float* __restrict__ Craw, float* __restrict__ Cinit,
    const float* __restrict__ dinv, int M, int N, int K)
{
    const int wave = threadIdx.x >> 5;
    const int lane = threadIdx.x & 31;
    const int half = lane >> 4;
    const int r    = lane & 15;
    const int m0 = blockIdx.x * 128 + wave * 16;
    const int n0 = blockIdx.y * 64;

    v8f c0 = {}; v8f c1 = {}; v8f c2 = {}; v8f c3 = {};

    const float* arow = A + (size_t)(m0 + r) * K + 2 * half;
    const float* bcol = B + (size_t)(2 * half) * N + n0 + r;

    for (int k0 = 0; k0 < K; k0 += 4) {
        v2f a = *(const v2f*)(arow + k0);
        const float* bp = bcol + (size_t)k0 * N;
        v2f b0 = { bp[ 0], bp[ 0 + N] };
        v2f b1 = { bp[16], bp[16 + N] };
        v2f b2 = { bp[32], bp[32 + N] };
        v2f b3 = { bp[48], bp[48 + N] };
        c0 = __builtin_amdgcn_wmma_f32_16x16x4_f32(false, a, false, b0, (short)0, c0, false, false);
        c1 = __builtin_amdgcn_wmma_f32_16x16x4_f32(false, a, false, b1, (short)0, c1, false, false);
        c2 = __builtin_amdgcn_wmma_f32_16x16x4_f32(false, a, false, b2, (short)0, c2, false, false);
        c3 = __builtin_amdgcn_wmma_f32_16x16x4_f32(false, a, false, b3, (short)0, c3, false, false);
    }

#pragma unroll
    for (int v = 0; v < 8; ++v) {
        int row = m0 + v + 8 * half;
        float di = dinv[row];
        float s  = di * di;
        size_t base = (size_t)row * N + n0 + r;
        float e0 = c0[v], e1 = c1[v], e2 = c2[v], e3 = c3[v];
        Craw [base     ] = e0;  Cinit[base     ] = e0 * s;
        Craw [base + 16] = e1;  Cinit[base + 16] = e1 * s;
        Craw [base + 32] = e2;  Cinit[base + 32] = e2 * s;
        Craw [base + 48] = e3;  Cinit[base + 48] = e3 * s;
    }
}

// ---------------------------------------------------------------------------
// Edge scatter:  agg[dst] += feat[src] * dinv[src]*dinv[dst]
// One thread per (edge, group of 4 features); float4 loads, f32 atomics.
// ---------------------------------------------------------------------------
__global__ __launch_bounds__(256) void edge_scatter_kernel(
    const long long* __restrict__ ei, const float* __restrict__ dinv,
    const float* __restrict__ feat, float* __restrict__ agg,
    int F, int lgGroups)
{
    int idx = blockIdx.x * blockDim.x + threadIdx.x;
    int e = idx >> lgGroups;
    int g = idx & ((1 << lgGroups) - 1);
    int s = (int)ei[e];
    int d = (int)ei[N_EDGES + e];
    float nrm = dinv[s] * dinv[d];
    const float4 v = *(const float4*)(feat + (size_t)s * F + g * 4);
    float* base = agg + (size_t)d * F + g * 4;
    atomicAdd(base + 0, v.x * nrm);
    atomicAdd(base + 1, v.y * nrm);
    atomicAdd(base + 2, v.z * nrm);
    atomicAdd(base + 3, v.w * nrm);
}

// ---------------------------------------------------------------------------
// Elementwise epilogues
// ---------------------------------------------------------------------------
__global__ void bias_relu_kernel(const float* __restrict__ agg,
                                 const float* __restrict__ b,
                                 float* __restrict__ h) {
    int i = blockIdx.x * blockDim.x + threadIdx.x;
    int f = i & (H_DIM - 1);
    float v = agg[i] + b[f];
    h[i] = v > 0.0f ? v : 0.0f;
}

// z = gnoise * exp(xu) + xs  (faithful to the reference's swapped decode args)
__global__ void z_kernel(const float* __restrict__ aggmu, const float* __restrict__ bmu,
                         const float* __restrict__ aggsg, const float* __restrict__ bsg,
                         const float* __restrict__ gnoise, float* __restrict__ z) {
    int i = blockIdx.x * blockDim.x + threadIdx.x;
    int f = i & (L_DIM - 1);
    float xu = aggmu[i] + bmu[f];
    float xs = aggsg[i] + bsg[f];
    z[i] = gnoise[i] * __expf(xu) + xs;
}

// ---------------------------------------------------------------------------
// out = sigmoid(z @ z^T), z: [8192, 64].  K=64 fits entirely in LDS, so each
// block stages its full A tile (128 rows) + B tile (64 rows) of z once with
// coalesced float4 loads, then runs the whole K loop out of LDS:
//   - kills the 8x per-wave redundancy on the B tile (L2 traffic 160KB->48KB
//     per 32KB of HBM output, keeping the kernel store-bandwidth-bound)
//   - row stride padded to 68 floats: 16B-aligned for b128 stores, and the
//     b64 fragment reads are bank-conflict-free (rows shift 4 banks; a wave's
//     32 lanes cover banks 4r+k .. 4r+k+3 disjointly)
// Block tile 128x64, grid (8192/64, 8192/128).
// ---------------------------------------------------------------------------
#define ZS 68                      // padded LDS row stride (floats)

__global__ __launch_bounds__(256) void zzt_sigmoid_kernel(
    const float* __restrict__ z, float* __restrict__ out)
{
    __shared__ float lz[(128 + 64) * ZS];     // 52224 bytes of the WGP's 320KB

    const int wave = threadIdx.x >> 5;
    const int lane = threadIdx.x & 31;
    const int half = lane >> 4;
    const int r    = lane & 15;
    const int m0 = blockIdx.y * 128;
    const int n0 = blockIdx.x * 64;

    // Cooperative stage: 256 threads, 16 threads per 64-float row.
    {
        const int rowIt = threadIdx.x >> 4;          // 0..15
        const int colf  = (threadIdx.x & 15) * 4;    // float offset in row
#pragma unroll
        for (int i = 0; i < 128; i += 16) {          // A tile rows
            float4 v = *(const float4*)(z + (size_t)(m0 + rowIt + i) * L_DIM + colf);
            *(float4*)&lz[(rowIt + i) * ZS + colf] = v;
        }
#pragma unroll
        for (int i = 0; i < 64; i += 16) {           // B tile rows
            float4 v = *(const float4*)(z + (size_t)(n0 + rowIt + i) * L_DIM + colf);
            *(float4*)&lz[(128 + rowIt + i) * ZS + colf] = v;
        }
    }
    __syncthreads();

    v8f c0 = {}; v8f c1 = {}; v8f c2 = {}; v8f c3 = {};

    const float* la = &lz[(wave * 16 + r) * ZS + 2 * half];
    const float* lb = &lz[(128 + r) * ZS + 2 * half];

#pragma unroll
    for (int k0 = 0; k0 < L_DIM; k0 += 4) {
        v2f a  = *(const v2f*)(la + k0);
        v2f b0 = *(const v2f*)(lb + k0);
        v2f b1 = *(const v2f*)(lb + 16 * ZS + k0);
        v2f b2 = *(const v2f*)(lb + 32 * ZS + k0);
        v2f b3 = *(const v2f*)(lb + 48 * ZS + k0);
        c0 = __builtin_amdgcn_wmma_f32_16x16x4_f32(false, a, false, b0, (short)0, c0, false, false);
        c1 = __builtin_amdgcn_wmma_f32_16x16x4_f32(false, a, false, b1, (short)0, c1, false, false);
        c2 = __builtin_amdgcn_wmma_f32_16x16x4_f32(false, a, false, b2, (short)0, c2, false, false);
        c3 = __builtin_amdgcn_wmma_f32_16x16x4_f32(false, a, false, b3, (short)0, c3, false, false);
    }

#pragma unroll
    for (int v = 0; v < 8; ++v) {
        size_t base = (size_t)(m0 + wave * 16 + v + 8 * half) * N_NODES + n0 + r;
        out[base     ] = 1.0f / (1.0f + __expf(-c0[v]));
        out[base + 16] = 1.0f / (1.0f + __expf(-c1[v]));
        out[base + 32] = 1.0f / (1.0f + __expf(-c2[v]));
        out[base + 48] = 1.0f / (1.0f + __expf(-c3[v]));
    }
}

// ---------------------------------------------------------------------------
extern "C" void kernel_launch(void* const* d_in, const int* in_sizes, int n_in,
                              void* d_out, int out_size, void* d_ws, size_t ws_size,
                              hipStream_t stream) {
    const float*     x    = (const float*)d_in[0];
    const long long* ei   = (const long long*)d_in[1];   // int64 edge_index [2, E]
    const float*     W1   = (const float*)d_in[2];
    const float*     b1   = (const float*)d_in[3];
    const float*     Wmu  = (const float*)d_in[4];
    const float*     bmu  = (const float*)d_in[5];
    const float*     Wsig = (const float*)d_in[6];
    const float*     bsig = (const float*)d_in[7];
    const float*     gn   = (const float*)d_in[8];
    float* out = (float*)d_out;

    float* ws     = (float*)d_ws;
    float* deg    = ws;                                  // 8192
    float* dinv   = deg    + N_NODES;                    // 8192
    float* xw     = dinv   + N_NODES;                    // 8192*256 raw X@W1
    float* agg1   = xw     + (size_t)N_NODES * H_DIM;    // 8192*256
    float* h      = agg1   + (size_t)N_NODES * H_DIM;    // 8192*256
    float* mu_raw = h      + (size_t)N_NODES * H_DIM;    // 8192*64
    float* aggmu  = mu_raw + (size_t)N_NODES * L_DIM;
    float* sg_raw = aggmu  + (size_t)N_NODES * L_DIM;
    float* aggsg  = sg_raw + (size_t)N_NODES * L_DIM;
    float* z      = aggsg  + (size_t)N_NODES * L_DIM;

    // normalization
    deg_init_kernel <<<N_NODES / 256, 256, 0, stream>>>(deg);
    deg_accum_kernel<<<N_EDGES / 256, 256, 0, stream>>>(ei, deg);
    dinv_kernel     <<<N_NODES / 256, 256, 0, stream>>>(deg, dinv);

    // layer 1: xw = x@W1 ; agg1 = xw*dinv^2 ; scatter edges ; h = relu(agg1+b1)
    wmma_gemm_f32_kernel<<<dim3(N_NODES / 128, H_DIM / 64), 256, 0, stream>>>(
        x, W1, xw, agg1, dinv, N_NODES, H_DIM, F_IN);
    edge_scatter_kernel<<<(N_EDGES * (H_DIM / 4)) / 256, 256, 0, stream>>>(
        ei, dinv, xw, agg1, H_DIM, 6);
    bias_relu_kernel<<<(N_NODES * H_DIM) / 256, 256, 0, stream>>>(agg1, b1, h);

    // layer 2, mu head
    wmma_gemm_f32_kernel<<<dim3(N_NODES / 128, L_DIM / 64), 256, 0, stream>>>(
        h, Wmu, mu_raw, aggmu, dinv, N_NODES, L_DIM, H_DIM);
    edge_scatter_kernel<<<(N_EDGES * (L_DIM / 4)) / 256, 256, 0, stream>>>(
        ei, dinv, mu_raw, aggmu, L_DIM, 4);

    // layer 2, logstd head
    wmma_gemm_f32_kernel<<<dim3(N_NODES / 128, L_DIM / 64), 256, 0, stream>>>(
        h, Wsig, sg_raw, aggsg, dinv, N_NODES, L_DIM, H_DIM);
    edge_scatter_kernel<<<(N_EDGES * (L_DIM / 4)) / 256, 256, 0, stream>>>(
        ei, dinv, sg_raw, aggsg, L_DIM, 4);

    // z = gnoise * exp(xu) + xs
    z_kernel<<<(N_NODES * L_DIM) / 256, 256, 0, stream>>>(aggmu, bmu, aggsg, bsig, gn, z);

    // out = sigmoid(z @ z^T)
    zzt_sigmoid_kernel<<<dim3(N_NODES / 64, N_NODES / 128), 256, 0, stream>>>(z, out);
}